// MultiboxLoss_7181185319051
// MI455X (gfx1250) — compile-verified
//
#include <hip/hip_runtime.h>

typedef float v2f __attribute__((ext_vector_type(2)));
typedef float v8f __attribute__((ext_vector_type(8)));

#define SENTINEL 0xFFFFFFFFu
#define INF_BITS 0x7F800000u

__device__ __forceinline__ float waveReduceSumF(float v) {
#pragma unroll
    for (int off = 16; off > 0; off >>= 1) v += __shfl_down(v, off, 32);
    return v;
}
__device__ __forceinline__ int waveReduceSumI(int v) {
#pragma unroll
    for (int off = 16; off > 0; off >>= 1) v += __shfl_down(v, off, 32);
    return v;
}

// ---------------------------------------------------------------------------
// K0: zero histograms + scalar state (S[0]=num_pos,S[1]=k,S[2]=b1,S[3]=r1,
// S[4]=b2,S[5]=r2,S[6]=t_bits,S[7]=r3)
// ---------------------------------------------------------------------------
__global__ void k_zero(unsigned* h1, unsigned* h2, unsigned* h3, unsigned* S) {
    for (int i = threadIdx.x; i < 2048; i += 256) { h1[i] = 0u; h2[i] = 0u; }
    for (int i = threadIdx.x; i < 1024; i += 256) h3[i] = 0u;
    if (threadIdx.x < 8) S[threadIdx.x] = 0u;
}

// ---------------------------------------------------------------------------
// K_main: per-anchor CE via WMMA row-sum of exp, loc L1 for positives,
// neg-CE scratch array, level-1 histogram, deterministic block partial sums.
// One wave handles 16 anchors (one 16x16x4-f32 WMMA tile chain).
// ---------------------------------------------------------------------------
__global__ void __launch_bounds__(256)
k_main(const float* __restrict__ pred_loc, const float* __restrict__ pred_clf,
       const float* __restrict__ target_loc, const int* __restrict__ target_cls,
       float* __restrict__ ce_neg, float* __restrict__ part_ce,
       float* __restrict__ part_l1, unsigned* __restrict__ hist1,
       unsigned* __restrict__ S, int N) {
    __shared__ unsigned shh[2048];
    __shared__ float sce[8], sl1[8];
    __shared__ int snp[8];
    const int tid = threadIdx.x;
    for (int i = tid; i < 2048; i += 256) shh[i] = 0u;
    __syncthreads();

    const int wave = tid >> 5, lane = tid & 31;
    const int base = (blockIdx.x * 8 + wave) * 16;
    float ceContrib = 0.0f, l1Contrib = 0.0f;
    int npContrib = 0;

    if (base < N) {  // wave-uniform: EXEC stays all-ones for WMMA
        const int hi = lane >> 4;  // 0: K={0,1} half, 1: K={2,3} half
        int aData = base + (lane & 15);
        if (aData >= N) aData = N - 1;  // tail clamp (N%128==0 in practice)
        const float* rowp = pred_clf + (size_t)aData * 21u;

        // Each lane loads its 10-11 classes per the 16x4 f32 A-matrix layout.
        float x[12];
        x[10] = -3.402823e38f;
        x[11] = -3.402823e38f;
#pragma unroll
        for (int ch = 0; ch < 5; ++ch) {
            const int c = ch * 4 + 2 * hi;  // <= 19
            x[2 * ch]     = rowp[c];
            x[2 * ch + 1] = rowp[c + 1];
        }
        if (hi == 0) x[10] = rowp[20];

        float mv = x[0];
#pragma unroll
        for (int i = 1; i < 12; ++i) mv = fmaxf(mv, x[i]);
        const float pmax = fmaxf(mv, __shfl_xor(mv, 16, 32));  // per-anchor max

        // Row-sum of exp via D = A x ones(4x16) + C, chained over 6 K-chunks.
        v8f acc = {0.f, 0.f, 0.f, 0.f, 0.f, 0.f, 0.f, 0.f};
        v2f bone;
        bone.x = 1.0f;
        bone.y = 1.0f;
#pragma unroll
        for (int ch = 0; ch < 6; ++ch) {
            v2f a;
            if (ch < 5) {
                a.x = __expf(x[2 * ch] - pmax);
                a.y = __expf(x[2 * ch + 1] - pmax);
            } else {  // classes 20 | pad
                a.x = (hi == 0) ? __expf(x[10] - pmax) : 0.0f;
                a.y = 0.0f;
            }
            acc = __builtin_amdgcn_wmma_f32_16x16x4_f32(
                false, a, false, bone, (short)0, acc, false, false);
        }

        // D layout: VGPR r, lanes 0-15 -> M=r ; lanes 16-31 -> M=8+r.
        // Writers: lanes 0-7 own anchors 0-7, lanes 16-23 own anchors 8-15.
        const int sel = lane & 7;
        float se = 0.0f;
#pragma unroll
        for (int i = 0; i < 8; ++i) se = (sel == i) ? acc[i] : se;

        const int mOut = (lane < 8) ? lane : (lane - 8);
        const float wmax = __shfl(pmax, mOut & 15, 32);  // full-EXEC shuffle
        const bool isW = (lane < 8) || (lane >= 16 && lane < 24);
        const int aOut = base + mOut;
        if (isW && aOut < N) {
            const float lse = __logf(se) + wmax;
            const int tgt = target_cls[aOut];
            const float xt = pred_clf[(size_t)aOut * 21u + tgt];
            const float cev = fmaxf(lse - xt, 0.0f);  // CE >= 0, kill -0
            if (tgt != 0) {  // positive anchor
                npContrib = 1;
                ceContrib = cev;
                const float4 pl = *reinterpret_cast<const float4*>(pred_loc + (size_t)aOut * 4u);
                const float4 tl = *reinterpret_cast<const float4*>(target_loc + (size_t)aOut * 4u);
                l1Contrib = fabsf(pl.x - tl.x) + fabsf(pl.y - tl.y) +
                            fabsf(pl.z - tl.z) + fabsf(pl.w - tl.w);
                ce_neg[aOut] = -1.0f;  // excluded from selection (sign bit set)
            } else {  // negative anchor
                ce_neg[aOut] = cev;
                const unsigned key = __float_as_uint(cev) & 0x7FFFFFFFu;
                atomicAdd(&shh[key >> 21], 1u);
            }
        }
    }

    // Deterministic block partials (fixed-order wave tree + thread-0 combine).
    const float rce = waveReduceSumF(ceContrib);
    const float rl1 = waveReduceSumF(l1Contrib);
    const int rnp = waveReduceSumI(npContrib);
    if (lane == 0) { sce[wave] = rce; sl1[wave] = rl1; snp[wave] = rnp; }
    __syncthreads();
    if (tid == 0) {
        float tc = 0.0f, tl = 0.0f;
        int tn = 0;
        for (int w = 0; w < 8; ++w) { tc += sce[w]; tl += sl1[w]; tn += snp[w]; }
        part_ce[blockIdx.x] = tc;
        part_l1[blockIdx.x] = tl;
        if (tn) atomicAdd(&S[0], (unsigned)tn);  // num_pos (int: deterministic)
    }
    for (int i = tid; i < 2048; i += 256) {
        const unsigned c = shh[i];
        if (c) atomicAdd(&hist1[i], c);
    }
}

// ---------------------------------------------------------------------------
// Radix-select scans (single thread; 2048 iters, trivial) and L2-resident
// refinement histograms over the 4.5 MB CE array.
// ---------------------------------------------------------------------------
__global__ void k_scan1(const unsigned* __restrict__ hist1, unsigned* __restrict__ S, int N) {
    const unsigned np = S[0];
    const unsigned nneg = (unsigned)N - np;
    unsigned k = 3u * np;
    if (k > nneg) k = nneg;
    S[1] = k;
    if (k == 0u) { S[2] = SENTINEL; S[3] = 0u; return; }
    unsigned cum = 0u;
    for (int b = 2047; b >= 0; --b) {
        const unsigned h = hist1[b];
        cum += h;
        if (cum >= k) { S[2] = (unsigned)b; S[3] = k - (cum - h); return; }
    }
    S[2] = SENTINEL; S[3] = 0u;
}

__global__ void __launch_bounds__(256)
k_hist2(const float* __restrict__ ce_neg, const unsigned* __restrict__ S,
        unsigned* __restrict__ hist2, int N) {
    __shared__ unsigned shh[2048];
    const unsigned b1 = S[2];
    for (int i = threadIdx.x; i < 2048; i += 256) shh[i] = 0u;
    __syncthreads();
    if (b1 != SENTINEL) {
        const int stride = gridDim.x * blockDim.x;
        for (int i = blockIdx.x * blockDim.x + threadIdx.x; i < N; i += stride) {
            const unsigned bits = __float_as_uint(ce_neg[i]);
            if ((bits >> 21) == b1) atomicAdd(&shh[(bits >> 10) & 0x7FFu], 1u);
        }
    }
    __syncthreads();
    for (int i = threadIdx.x; i < 2048; i += 256) {
        const unsigned c = shh[i];
        if (c) atomicAdd(&hist2[i], c);
    }
}

__global__ void k_scan2(const unsigned* __restrict__ hist2, unsigned* __restrict__ S) {
    if (S[2] == SENTINEL) { S[4] = SENTINEL; S[5] = 0u; return; }
    const unsigned r = S[3];
    unsigned cum = 0u;
    for (int b = 2047; b >= 0; --b) {
        const unsigned h = hist2[b];
        cum += h;
        if (cum >= r) { S[4] = (unsigned)b; S[5] = r - (cum - h); return; }
    }
    S[4] = SENTINEL; S[5] = 0u;
}

__global__ void __launch_bounds__(256)
k_hist3(const float* __restrict__ ce_neg, const unsigned* __restrict__ S,
        unsigned* __restrict__ hist3, int N) {
    __shared__ unsigned shh[1024];
    const unsigned b2 = S[4];
    for (int i = threadIdx.x; i < 1024; i += 256) shh[i] = 0u;
    __syncthreads();
    if (b2 != SENTINEL) {
        const unsigned pfx = (S[2] << 11) | b2;
        const int stride = gridDim.x * blockDim.x;
        for (int i = blockIdx.x * blockDim.x + threadIdx.x; i < N; i += stride) {
            const unsigned bits = __float_as_uint(ce_neg[i]);
            if ((bits >> 10) == pfx) atomicAdd(&shh[bits & 0x3FFu], 1u);
        }
    }
    __syncthreads();
    for (int i = threadIdx.x; i < 1024; i += 256) {
        const unsigned c = shh[i];
        if (c) atomicAdd(&hist3[i], c);
    }
}

__global__ void k_scan3(const unsigned* __restrict__ hist3, unsigned* __restrict__ S) {
    const unsigned b2 = S[4];
    if (b2 == SENTINEL) { S[6] = INF_BITS; S[7] = 0u; return; }
    const unsigned r = S[5];
    unsigned cum = 0u;
    for (int b = 1023; b >= 0; --b) {
        const unsigned h = hist3[b];
        cum += h;
        if (cum >= r) {
            S[6] = (S[2] << 21) | (b2 << 10) | (unsigned)b;  // exact k-th value
            S[7] = r - (cum - h);                            // ties of t to take
            return;
        }
    }
    S[6] = INF_BITS; S[7] = 0u;
}

// Sum of CE values strictly greater than threshold t (deterministic partials).
__global__ void __launch_bounds__(256)
k_sumgt(const float* __restrict__ ce_neg, const unsigned* __restrict__ S,
        float* __restrict__ part_sum, int N) {
    __shared__ float ssum[8];
    const float t = __uint_as_float(S[6]);
    float local = 0.0f;
    const int stride = gridDim.x * blockDim.x;
    for (int i = blockIdx.x * blockDim.x + threadIdx.x; i < N; i += stride) {
        const float v = ce_neg[i];  // positives stored as -1.0f: never > t
        if (v > t) local += v;
    }
    const float r = waveReduceSumF(local);
    const int wave = threadIdx.x >> 5, lane = threadIdx.x & 31;
    if (lane == 0) ssum[wave] = r;
    __syncthreads();
    if (threadIdx.x == 0) {
        float tsum = 0.0f;
        for (int w = 0; w < 8; ++w) tsum += ssum[w];
        part_sum[blockIdx.x] = tsum;
    }
}

// Final deterministic reduction of all partials + scalar loss math.
__global__ void __launch_bounds__(256)
k_final(const float* __restrict__ part_ce, const float* __restrict__ part_l1,
        const float* __restrict__ part_sum, const unsigned* __restrict__ S,
        float* __restrict__ out, int NB1, int NB2) {
    __shared__ float sh[8];
    __shared__ float res[3];
    const int tid = threadIdx.x, wave = tid >> 5, lane = tid & 31;
    const float* arrs[3] = {part_ce, part_l1, part_sum};
    const int lens[3] = {NB1, NB1, NB2};
    for (int a = 0; a < 3; ++a) {
        float acc = 0.0f;
        for (int i = tid; i < lens[a]; i += 256) acc += arrs[a][i];
        acc = waveReduceSumF(acc);
        if (lane == 0) sh[wave] = acc;
        __syncthreads();
        if (tid == 0) {
            float s = 0.0f;
            for (int w = 0; w < 8; ++w) s += sh[w];
            res[a] = s;
        }
        __syncthreads();
    }
    if (tid == 0) {
        const float pce = res[0], pl1 = res[1], sum_gt = res[2];
        const float np = (float)S[0];
        const float kf = (float)S[1];
        const unsigned tb = S[6];
        const float tv = (tb == INF_BITS) ? 0.0f : __uint_as_float(tb);
        const float topk = sum_gt + (float)S[7] * tv;
        out[0] = (pce + topk) / (np + kf);  // loss_cls
        out[1] = pl1 / np;                  // loss_loc
    }
}

// ---------------------------------------------------------------------------
extern "C" void kernel_launch(void* const* d_in, const int* in_sizes, int n_in,
                              void* d_out, int out_size, void* d_ws, size_t ws_size,
                              hipStream_t stream) {
    (void)n_in; (void)out_size; (void)ws_size;
    const float* pred_loc   = (const float*)d_in[0];
    const float* pred_clf   = (const float*)d_in[1];
    const float* target_loc = (const float*)d_in[2];
    const int*   target_cls = (const int*)d_in[3];
    const int N = in_sizes[3];             // B*A anchors
    const int NB1 = (N + 127) / 128;       // k_main blocks (128 anchors each)
    const int NB2 = 1024;                  // sum-pass blocks

    float* ce_neg   = (float*)d_ws;        // N floats
    float* part_ce  = ce_neg + N;          // NB1
    float* part_l1  = part_ce + NB1;       // NB1
    float* part_sum = part_l1 + NB1;       // NB2
    unsigned* hist1 = (unsigned*)(part_sum + NB2);  // 2048
    unsigned* hist2 = hist1 + 2048;                 // 2048
    unsigned* hist3 = hist2 + 2048;                 // 1024
    unsigned* S     = hist3 + 1024;                 // 8 scalars
    float* out = (float*)d_out;

    k_zero <<<1, 256, 0, stream>>>(hist1, hist2, hist3, S);
    k_main <<<NB1, 256, 0, stream>>>(pred_loc, pred_clf, target_loc, target_cls,
                                     ce_neg, part_ce, part_l1, hist1, S, N);
    k_scan1<<<1, 1, 0, stream>>>(hist1, S, N);
    k_hist2<<<1024, 256, 0, stream>>>(ce_neg, S, hist2, N);
    k_scan2<<<1, 1, 0, stream>>>(hist2, S);
    k_hist3<<<1024, 256, 0, stream>>>(ce_neg, S, hist3, N);
    k_scan3<<<1, 1, 0, stream>>>(hist3, S);
    k_sumgt<<<NB2, 256, 0, stream>>>(ce_neg, S, part_sum, N);
    k_final<<<1, 256, 0, stream>>>(part_ce, part_l1, part_sum, S, out, NB1, NB2);
}